// StructuralProbeWithRegularization_25958782337133
// MI455X (gfx1250) — compile-verified
//
#include <hip/hip_runtime.h>

// StructuralProbe distances: D = ||p_i||^2 + ||p_j||^2 - 2 p_i.p_j + eps*I
// where P = E @ W  (E:[8192,1024] f32, W:[1024,128] f32)
//
// P is stored in a WMMA-fragment-native tiled layout:
//   float index of P[m][n] = ((m>>4)*32 + (n>>2))*64 + ((m&15) + 16*((n>>1)&1))*2 + (n&1)
// i.e. [rowTile][kChunk][lane]{2 floats}, so that stage-2 fragment loads and
// LDS staging copies are fully coalesced.

typedef float v2f __attribute__((ext_vector_type(2)));
typedef float v4f __attribute__((ext_vector_type(4)));
typedef float v8f __attribute__((ext_vector_type(8)));

#define NPTS 8192
#define MDIM 1024
#define RANK 128
#define TILE_FLOATS 2048   // one 16-row tile: 32 chunks * 32 lanes * 2 floats

// ---------------------------------------------------------------------------
// Stage 1: P = E @ W   (one 16x16 output tile per wave, WMMA f32 16x16x4)
// grid: (NPTS/16)*(RANK/16) = 4096 tiles, 8 waves/block -> 512 blocks x 256
// ---------------------------------------------------------------------------
__global__ __launch_bounds__(256)
void probe_project_kernel(const float* __restrict__ E,
                          const float* __restrict__ W,
                          float* __restrict__ P) {
    const int lane = threadIdx.x & 31;
    const int wave = threadIdx.x >> 5;
    const int l15  = lane & 15;
    const int hi   = lane >> 4;            // 0 or 1

    const int tile = blockIdx.x * 8 + wave;          // 0 .. 4095
    const int tileRow = tile >> 3;                   // RANK/16 = 8 col tiles
    const int tileCol = tile & 7;
    const int m0 = tileRow * 16;
    const int n0 = tileCol * 16;

    v8f acc = {};
    const float* eRow = E + (size_t)(m0 + l15) * MDIM;

    for (int kk = 0; kk < MDIM; kk += 4) {
        const int k = kk + 2 * hi;
        // A: 16x4 tile of E, lane holds row m0+l15, K = k, k+1
        v2f a = *(const v2f*)(eRow + k);
        // B: 4x16 tile of W (row-major [MDIM,RANK]), lane holds col n0+l15
        v2f b;
        b.x = W[(size_t)k * RANK + n0 + l15];
        b.y = W[(size_t)(k + 1) * RANK + n0 + l15];
        acc = __builtin_amdgcn_wmma_f32_16x16x4_f32(
            false, a, false, b, (short)0, acc, false, false);
    }

    // D layout: VGPR r -> M = m0 + r + 8*hi, N = n0 + l15.
    // Scatter into the tiled fragment layout (m0 is 16-aligned -> tile fixed).
    const int n     = n0 + l15;
    const int chunk = n >> 2;
    const int nh    = (n >> 1) & 1;
    const int slot  = n & 1;
    float* pb = P + (size_t)tileRow * TILE_FLOATS + chunk * 64 + slot;
    #pragma unroll
    for (int r = 0; r < 8; ++r) {
        pb[(size_t)(r + 8 * hi + 16 * nh) * 2] = acc[r];
    }
}

// ---------------------------------------------------------------------------
// Stage 1b: sq[i] = sum_k P[i,k]^2   (P in tiled layout)
// ---------------------------------------------------------------------------
__global__ __launch_bounds__(256)
void probe_rownorm_kernel(const float* __restrict__ P,
                          float* __restrict__ sq) {
    const int row = blockIdx.x * blockDim.x + threadIdx.x;
    if (row >= NPTS) return;
    const int tile = row >> 4;
    const int l    = row & 15;
    const float* base = P + (size_t)tile * TILE_FLOATS;
    float s = 0.f;
    #pragma unroll 4
    for (int chunk = 0; chunk < 32; ++chunk) {
        v2f x0 = *(const v2f*)(base + chunk * 64 + l * 2);          // n = 4c+0,1
        v2f x1 = *(const v2f*)(base + chunk * 64 + (l + 16) * 2);   // n = 4c+2,3
        s += x0.x * x0.x + x0.y * x0.y + x1.x * x1.x + x1.y * x1.y;
    }
    sq[row] = s;
}

// ---------------------------------------------------------------------------
// Stage 2: D[i,j] = sq[i] + sq[j] - 2*dot(P_i, P_j) + (i==j)*1e-8
// Block: 256 threads = 8 waves (2x4) -> 128x256 output tile.
// K staged through LDS in 4 chunk-groups of 8 chunks (K=32 each):
//   A panel 8 tiles * 2KB = 16KB, B panel 16 tiles * 2KB = 32KB -> 48KB LDS.
// Each wave: 64x64 tile = 4x4 subtiles, fragments via conflict-free ds_load_b64.
// grid: (8192/128, 8192/256) = (64, 32)
// ---------------------------------------------------------------------------
__global__ __launch_bounds__(256)
void probe_dist_kernel(const float* __restrict__ P,
                       const float* __restrict__ sq,
                       float* __restrict__ out) {
    __shared__ float lds[12288];           // [0,4096) = A, [4096,12288) = B

    const int tid  = threadIdx.x;
    const int lane = tid & 31;
    const int wave = tid >> 5;
    const int l15  = lane & 15;
    const int hi   = lane >> 4;

    const int wr = wave & 1;               // 0..1
    const int wc = wave >> 1;              // 0..3
    const int rowTile0 = blockIdx.x * 8;   // 128 rows  = 8 tiles
    const int colTile0 = blockIdx.y * 16;  // 256 cols  = 16 tiles
    const int row0 = rowTile0 * 16 + wr * 64;
    const int col0 = colTile0 * 16 + wc * 64;

    v8f acc[4][4] = {};

    for (int cb = 0; cb < 4; ++cb) {       // K chunk-group (8 chunks of K=4)
        __syncthreads();
        // Stage 48KB: 3072 float4 (A: 8 segs * 128, B: 16 segs * 128),
        // each segment = one row-tile's 8 chunks = 512 floats, contiguous
        // both in global (tiled layout) and in LDS.
        for (int i = tid; i < 3072; i += 256) {
            int seg, off, gtile;
            float* dst;
            if (i < 1024) {
                seg = i >> 7; off = i & 127;
                gtile = rowTile0 + seg;
                dst = lds + seg * 512 + off * 4;
            } else {
                const int j = i - 1024;
                seg = j >> 7; off = j & 127;
                gtile = colTile0 + seg;
                dst = lds + 4096 + seg * 512 + off * 4;
            }
            const float* src = P + (size_t)gtile * TILE_FLOATS + cb * 512 + off * 4;
            *(v4f*)dst = *(const v4f*)src;
            if (cb < 3) __builtin_prefetch(src + 512, 0, 0);  // next chunk-group
        }
        __syncthreads();

        #pragma unroll
        for (int cl = 0; cl < 8; ++cl) {   // chunk within group (K step of 4)
            v2f a[4], b[4];
            #pragma unroll
            for (int i2 = 0; i2 < 4; ++i2)
                a[i2] = *(const v2f*)(lds + ((wr * 4 + i2) * 8 + cl) * 64 + lane * 2);
            #pragma unroll
            for (int j2 = 0; j2 < 4; ++j2)
                b[j2] = *(const v2f*)(lds + 4096 + ((wc * 4 + j2) * 8 + cl) * 64 + lane * 2);

            #pragma unroll
            for (int i2 = 0; i2 < 4; ++i2)
                #pragma unroll
                for (int j2 = 0; j2 < 4; ++j2)
                    acc[i2][j2] = __builtin_amdgcn_wmma_f32_16x16x4_f32(
                        false, a[i2], false, b[j2], (short)0, acc[i2][j2],
                        false, false);
        }
    }

    // Epilogue: norms + eps on diagonal, write fp32 result.
    #pragma unroll
    for (int j = 0; j < 4; ++j) {
        const int col = col0 + 16 * j + l15;
        const float sqc = sq[col];
        #pragma unroll
        for (int i = 0; i < 4; ++i) {
            #pragma unroll
            for (int r = 0; r < 8; ++r) {
                const int row = row0 + 16 * i + r + 8 * hi;
                float v = sq[row] + sqc - 2.0f * acc[i][j][r];
                if (row == col) v += 1e-8f;
                out[(size_t)row * NPTS + col] = v;
            }
        }
    }
}

// ---------------------------------------------------------------------------
extern "C" void kernel_launch(void* const* d_in, const int* in_sizes, int n_in,
                              void* d_out, int out_size, void* d_ws, size_t ws_size,
                              hipStream_t stream) {
    const float* E = (const float*)d_in[0];   // [8192, 1024]
    const float* W = (const float*)d_in[1];   // [1024, 128]
    float* out = (float*)d_out;               // [8192, 8192]

    float* P  = (float*)d_ws;                 // [8192, 128] tiled (4 MB)
    float* sq = P + (size_t)NPTS * RANK;      // [8192]

    // Stage 1: projection (4096 wave-tiles, 8 waves per block)
    probe_project_kernel<<<dim3(512), dim3(256), 0, stream>>>(E, W, P);

    // Stage 1b: row squared norms
    probe_rownorm_kernel<<<dim3(NPTS / 256), dim3(256), 0, stream>>>(P, sq);

    // Stage 2: pairwise distances (write-bandwidth bound, 256 MB out)
    probe_dist_kernel<<<dim3(NPTS / 128, NPTS / 256), dim3(256), 0, stream>>>(P, sq, out);
}